// SpatialLearnedEmbeddings_14010183320158
// MI455X (gfx1250) — compile-verified
//
#include <hip/hip_runtime.h>
#include <stdint.h>

// SpatialLearnedEmbeddings: y[b, f*C + c] = sum_hw x[b,c,h,w] * kernel[f,c,h,w]
// Per-channel GEMM X_c[256x1024] * K_c^T[1024x16] -> Y_c[256x16], fp32 WMMA.
//
// Data movement (MI455X-specific):
//  - global_load_async_to_lds_b128 streams coalesced 128B-line-aligned chunks
//    of x / kernel into LDS (double-buffered, tracked with ASYNCcnt).
//  - WMMA fragments are read from LDS with ds_load_b64, conflict-free via
//    row padding (stride 36 floats: bank = (36r+o) mod 64, 9 coprime 16).
//  - V_WMMA_F32_16X16X4_F32 accumulates in fp32.

typedef float v2f __attribute__((ext_vector_type(2)));
typedef float v8f __attribute__((ext_vector_type(8)));

#define B_DIM 256
#define F_DIM 16
#define C_DIM 256
#define HW_DIM 1024

#define WPB 4                     // waves per block (128 threads)
#define MT  4                     // 16-row M-subtiles per wave (64 batches)
#define KC  32                    // K-chunk in floats (128 B per row)
#define NCHUNK (HW_DIM / KC)      // 32 chunks
#define PADF 4                    // LDS row padding (floats) -> bank-conflict-free
#define STRIDEF (KC + PADF)       // 36 floats per LDS row
#define A_TILE_F (B_DIM * STRIDEF)
#define B_TILE_F (F_DIM * STRIDEF)
#define BUF_F (A_TILE_F + B_TILE_F)
#define SMEM_BYTES (2 * BUF_F * (int)sizeof(float))   // 78,336 B < 320 KB

#define S_WAIT_ASYNCCNT(n) asm volatile("s_wait_asynccnt " #n ::: "memory")

__device__ __forceinline__ void async_b128(unsigned lds_byte, const float* g) {
    unsigned long long ga = (unsigned long long)(const void*)g;
    // VDST = per-lane LDS byte address, VADDR = 64-bit global address, saddr off
    asm volatile("global_load_async_to_lds_b128 %0, %1, off"
                 :: "v"(lds_byte), "v"(ga) : "memory");
}

__global__ __launch_bounds__(WPB * 32)
void sle_wmma_async_kernel(const float* __restrict__ x,
                           const float* __restrict__ ker,
                           float* __restrict__ out)
{
    extern __shared__ float smem[];

    const int tid  = threadIdx.x;
    const int w    = tid >> 5;        // wave id 0..3
    const int lane = tid & 31;
    const int c    = blockIdx.x;      // channel

    const int half = lane >> 4;       // WMMA K-half: 0 -> K{0,1}, 1 -> K{2,3}
    const int l    = lane & 15;       // M row (A) / N column == filter (B)

    // Loader mapping: one b128 instruction covers 4 complete rows x 128 B
    const int rq = lane >> 3;         // row within quad (0..3)
    const int ju = lane & 7;          // 16-byte unit within row (0..7)

    const size_t rowstride = (size_t)C_DIM * HW_DIM;
    const float* xg = x   + (size_t)c * HW_DIM + (size_t)ju * 4;
    const float* kg = ker + (size_t)c * HW_DIM + (size_t)ju * 4;

    auto load_chunk = [&](int ck, int buf) {
        const size_t k0 = (size_t)ck * KC;
        float* Ab = smem + buf * BUF_F;
        float* Bb = Ab + A_TILE_F;
        // A: this wave loads its own 64 batch rows (16 instructions)
#pragma unroll
        for (int it = 0; it < 16; ++it) {
            const int row = w * 64 + it * 4 + rq;
            const unsigned lds = (unsigned)(uintptr_t)(Ab + row * STRIDEF + ju * 4);
            async_b128(lds, xg + (size_t)row * rowstride + k0);
        }
        // B: 16 filter rows split across the 4 waves (1 instruction)
        {
            const int row = w * 4 + rq;
            const unsigned lds = (unsigned)(uintptr_t)(Bb + row * STRIDEF + ju * 4);
            async_b128(lds, kg + (size_t)row * rowstride + k0);
        }
    };

    v8f acc[MT] = {};   // fp32 16x16 accumulators, 8 VGPRs each

    load_chunk(0, 0);

    for (int ck = 0; ck < NCHUNK; ++ck) {
        const int buf = ck & 1;
        if (ck + 1 < NCHUNK) {
            load_chunk(ck + 1, buf ^ 1);     // prefetch next chunk
            S_WAIT_ASYNCCNT(17);             // own chunk-ck loads complete
        } else {
            S_WAIT_ASYNCCNT(0);
        }
        __syncthreads();                     // all waves' chunk-ck data in LDS

        const float* Ab = smem + buf * BUF_F;
        const float* Bb = Ab + A_TILE_F;
#pragma unroll
        for (int kk = 0; kk < KC; kk += 4) {
            const v2f bfrag = *(const v2f*)(Bb + l * STRIDEF + kk + 2 * half);
#pragma unroll
            for (int t = 0; t < MT; ++t) {
                const v2f afrag =
                    *(const v2f*)(Ab + (w * 64 + t * 16 + l) * STRIDEF + kk + 2 * half);
                // (neg_a, A, neg_b, B, c_mod, C, reuse_a, reuse_b)
                acc[t] = __builtin_amdgcn_wmma_f32_16x16x4_f32(
                    false, afrag, false, bfrag, (short)0, acc[t], false, false);
            }
        }
        __syncthreads();                     // buffer free before overwrite
    }

    // D layout: VGPR v -> M = v + 8*half, N = l. out[b*(F*C) + f*C + c]
#pragma unroll
    for (int t = 0; t < MT; ++t) {
#pragma unroll
        for (int v = 0; v < 8; ++v) {
            const int b = w * 64 + t * 16 + v + half * 8;
            out[(size_t)b * (F_DIM * C_DIM) + (size_t)l * C_DIM + (size_t)c] = acc[t][v];
        }
    }
}

extern "C" void kernel_launch(void* const* d_in, const int* in_sizes, int n_in,
                              void* d_out, int out_size, void* d_ws, size_t ws_size,
                              hipStream_t stream) {
    const float* x   = (const float*)d_in[0];   // [B, C, H, W] fp32
    const float* ker = (const float*)d_in[1];   // [F, C, H, W] fp32
    float* out       = (float*)d_out;           // [B, F*C] fp32

    dim3 grid(C_DIM);          // one workgroup per channel
    dim3 block(WPB * 32);      // 128 threads = 4 wave32
    sle_wmma_async_kernel<<<grid, block, SMEM_BYTES, stream>>>(x, ker, out);
}